// NodeNetwork_69526930588457
// MI455X (gfx1250) — compile-verified
//
#include <hip/hip_runtime.h>
#include <math.h>

// ---------------------------------------------------------------------------
// Problem constants (match reference)
// ---------------------------------------------------------------------------
#define NNODES 50000
#define NEDGES 800000
#define FEA    64      // ELEM_FEA_LEN
#define EMB    92      // ELEM_EMB_LEN
#define NBR    41      // NBR_FEA_LEN
#define KCAT   169     // 2*FEA + NBR
#define KPAD   172     // padded to multiple of 4 for 16x16x4 WMMA k-steps
#define KMAIN  168     // unguarded k-steps; last step (168..171) guarded
#define N2F    128     // 2*FEA
#define NGRAPH 4
#define BN_EPS 1e-5f

#define TILE_M 64      // edges per block (4 row-tiles of 16)
#define ASTR   173     // LDS stride for A tile (gcd(45,64)=1 -> conflict-free)
#define DSTR   132     // LDS stride for D tile

typedef __attribute__((ext_vector_type(2))) float v2f;
typedef __attribute__((ext_vector_type(8))) float v8f;

// Full-precision CDNA5 matrix op: D(16x16,f32) = A(16x4,f32) x B(4x16,f32) + C
__device__ __forceinline__ v8f wmma_f32_16x16x4(v2f a, v2f b, v8f c) {
  return __builtin_amdgcn_wmma_f32_16x16x4_f32(
      /*neg_a=*/false, a, /*neg_b=*/false, b,
      /*c_mod=*/(short)0, c, /*reuse_a=*/false, /*reuse_b=*/false);
}

__device__ __forceinline__ float sigmoidf_(float x) {
  return 1.0f / (1.0f + __expf(-x));
}
__device__ __forceinline__ float softplusf_(float x) {
  return (x > 20.0f) ? x : log1pf(__expf(x));
}

// ---------------------------------------------------------------------------
// Kernel 1: x = atom_fea @ emb_W + emb_b   (50000 x 92 x 64)  (<1% of FLOPs)
// 128 threads = 4 waves; block = 16 rows; each wave owns one 16-col tile.
// ---------------------------------------------------------------------------
__global__ __launch_bounds__(128) void embed_kernel(
    const float* __restrict__ atom, const float* __restrict__ W,
    const float* __restrict__ bias, float* __restrict__ x) {
  __shared__ float As[16 * 93];
  const int tid  = threadIdx.x;
  const int row0 = blockIdx.x * 16;

  for (int idx = tid; idx < 16 * EMB; idx += 128) {
    int r = idx / EMB, k = idx - r * EMB;
    As[r * 93 + k] = atom[(size_t)(row0 + r) * EMB + k];
  }
  __syncthreads();

  const int lane = tid & 31;
  const int wave = tid >> 5;           // 0..3 -> N tile
  const int m    = lane & 15;
  const int kh   = (lane >> 4) << 1;   // 0 or 2
  const int col  = wave * 16 + m;

  v8f acc = {};
  for (int k0 = 0; k0 < EMB; k0 += 4) {  // 92 = 23*4, no remainder
    int ka = k0 + kh;
    v2f a, b;
    a.x = As[m * 93 + ka];
    a.y = As[m * 93 + ka + 1];
    b.x = W[ka * FEA + col];
    b.y = W[(ka + 1) * FEA + col];
    acc = wmma_f32_16x16x4(a, b, acc);
  }

  const float bv    = bias[col];
  const int   rbase = (lane >> 4) * 8;
#pragma unroll
  for (int r = 0; r < 8; ++r)
    x[(size_t)(row0 + rbase + r) * FEA + col] = acc[r] + bv;
}

// ---------------------------------------------------------------------------
// Shared helper: build the TILE_M-edge x KPAD concatenated feature tile in LDS.
// ---------------------------------------------------------------------------
__device__ __forceinline__ void load_edge_tile(
    float* As /* TILE_M*ASTR */, int* sidx, int* nidx,
    const float* __restrict__ x, const float* __restrict__ nbrf,
    const int* __restrict__ selfi, const int* __restrict__ nbri,
    long long e0, int tid) {
  if (tid < TILE_M)           sidx[tid]          = selfi[e0 + tid];
  else if (tid < 2 * TILE_M)  nidx[tid - TILE_M] = nbri[e0 + tid - TILE_M];
  __syncthreads();
  for (int idx = tid; idx < TILE_M * KPAD; idx += 256) {
    int r = idx / KPAD, f = idx - r * KPAD;
    float v;
    if (f < FEA)            v = x[(size_t)sidx[r] * FEA + f];
    else if (f < 2 * FEA)   v = x[(size_t)nidx[r] * FEA + (f - FEA)];
    else {
      int nf = f - 2 * FEA;                       // padded slots 169..171 -> 0
      v = (nf < NBR) ? nbrf[(e0 + r) * NBR + nf] : 0.0f;
    }
    As[r * ASTR + f] = v;
  }
  __syncthreads();
}

// ---------------------------------------------------------------------------
// Shared helper: one wave computes a 64x16 strip (4 row-tiles) of cat @ Wl.
// One B fragment fetched per k-step is reused by 4 WMMAs -> 4x B reuse.
// ---------------------------------------------------------------------------
__device__ __forceinline__ void edge_gemm_strip4(
    const float* As, const float* __restrict__ Wl,
    int m, int kh, int col, v8f acc[4]) {
  // main loop: k < 168, no bounds checks on W rows
  for (int k0 = 0; k0 < KMAIN; k0 += 4) {
    const int ka = k0 + kh;
    v2f b;
    b.x = Wl[ka * N2F + col];
    b.y = Wl[(ka + 1) * N2F + col];
#pragma unroll
    for (int t = 0; t < 4; ++t) {
      v2f a;
      a.x = As[(t * 16 + m) * ASTR + ka];
      a.y = As[(t * 16 + m) * ASTR + ka + 1];
      acc[t] = wmma_f32_16x16x4(a, b, acc[t]);
    }
  }
  // guarded tail step: k rows 168..171; W has only 169 rows (A pad is zero)
  {
    const int ka = KMAIN + kh;
    v2f b;
    b.x = (ka     < KCAT) ? Wl[ka * N2F + col]       : 0.0f;
    b.y = (ka + 1 < KCAT) ? Wl[(ka + 1) * N2F + col] : 0.0f;
#pragma unroll
    for (int t = 0; t < 4; ++t) {
      v2f a;
      a.x = As[(t * 16 + m) * ASTR + ka];
      a.y = As[(t * 16 + m) * ASTR + ka + 1];
      acc[t] = wmma_f32_16x16x4(a, b, acc[t]);
    }
  }
}

// ---------------------------------------------------------------------------
// Kernel 2 (per layer, pass 1): edge GEMM + per-column sum/sumsq for BN1.
// 256 threads = 8 waves; block = 64 edges x 128 cols.
// ---------------------------------------------------------------------------
__global__ __launch_bounds__(256) void edge_pass1(
    const float* __restrict__ x, const float* __restrict__ nbrf,
    const int* __restrict__ selfi, const int* __restrict__ nbri,
    const float* __restrict__ Wl, const float* __restrict__ bl,
    float* __restrict__ gstat /* [0..127]=sum, [128..255]=sumsq */) {
  __shared__ float As[TILE_M * ASTR];
  __shared__ int   sidx[TILE_M], nidx[TILE_M];
  __shared__ float bsum[N2F], bsq[N2F];

  const int tid = threadIdx.x;
  const long long e0 = (long long)blockIdx.x * TILE_M;
  if (tid < N2F) { bsum[tid] = 0.0f; bsq[tid] = 0.0f; }
  load_edge_tile(As, sidx, nidx, x, nbrf, selfi, nbri, e0, tid);

  const int lane = tid & 31;
  const int wave = tid >> 5;
  const int m    = lane & 15;
  const int kh   = (lane >> 4) << 1;
  const int col  = wave * 16 + m;

  v8f acc[4] = {};
  edge_gemm_strip4(As, Wl, m, kh, col, acc);

  const float bv = bl[col];
  float s = 0.0f, q = 0.0f;
#pragma unroll
  for (int t = 0; t < 4; ++t) {
#pragma unroll
    for (int r = 0; r < 8; ++r) {
      float v = acc[t][r] + bv;
      s += v; q += v * v;
    }
  }
  // lanes L and L^16 hold the same column -> combine, then one add per col
  s += __shfl_xor(s, 16, 32);
  q += __shfl_xor(q, 16, 32);
  if (lane < 16) { atomicAdd(&bsum[col], s); atomicAdd(&bsq[col], q); }
  __syncthreads();
  if (tid < N2F) atomicAdd(&gstat[tid], bsum[tid]);
  else           atomicAdd(&gstat[tid], bsq[tid - N2F]);
}

// ---------------------------------------------------------------------------
// Kernel 3: fold BN1 stats into per-column scale/shift.
// ---------------------------------------------------------------------------
__global__ void bn1_finalize(const float* __restrict__ gstat,
                             const float* __restrict__ g,
                             const float* __restrict__ b,
                             float* __restrict__ out /* scale[128],shift[128] */) {
  const int t = threadIdx.x;  // 128 threads
  const float invE = 1.0f / (float)NEDGES;
  float mean = gstat[t] * invE;
  float var  = gstat[N2F + t] * invE - mean * mean;
  float sc   = g[t] * rsqrtf(var + BN_EPS);
  out[t]       = sc;
  out[N2F + t] = b[t] - mean * sc;
}

// ---------------------------------------------------------------------------
// Kernel 4 (per layer, pass 2): recompute GEMM, BN affine, sigmoid*softplus,
// scatter-add messages into summed[self_idx].
// ---------------------------------------------------------------------------
__global__ __launch_bounds__(256) void edge_pass2(
    const float* __restrict__ x, const float* __restrict__ nbrf,
    const int* __restrict__ selfi, const int* __restrict__ nbri,
    const float* __restrict__ Wl, const float* __restrict__ bl,
    const float* __restrict__ bn1 /* scale[128],shift[128] */,
    float* __restrict__ summed) {
  __shared__ float As[TILE_M * ASTR];
  __shared__ float Dt[TILE_M * DSTR];  // normalized 64x128 tile
  __shared__ int   sidx[TILE_M], nidx[TILE_M];

  const int tid = threadIdx.x;
  const long long e0 = (long long)blockIdx.x * TILE_M;
  load_edge_tile(As, sidx, nidx, x, nbrf, selfi, nbri, e0, tid);

  const int lane = tid & 31;
  const int wave = tid >> 5;
  const int m    = lane & 15;
  const int kh   = (lane >> 4) << 1;
  const int col  = wave * 16 + m;

  v8f acc[4] = {};
  edge_gemm_strip4(As, Wl, m, kh, col, acc);

  const float bv = bl[col];
  const float sc = bn1[col], sh = bn1[N2F + col];
  const int rbase = (lane >> 4) * 8;
#pragma unroll
  for (int t = 0; t < 4; ++t)
#pragma unroll
    for (int r = 0; r < 8; ++r)
      Dt[(t * 16 + rbase + r) * DSTR + col] = (acc[t][r] + bv) * sc + sh;
  __syncthreads();

  // msg = sigmoid(filt) * softplus(core); scatter-add by self_idx
  for (int idx = tid; idx < TILE_M * FEA; idx += 256) {
    int r = idx >> 6, f = idx & 63;
    float msg = sigmoidf_(Dt[r * DSTR + f]) * softplusf_(Dt[r * DSTR + f + FEA]);
    atomicAdd(&summed[(size_t)sidx[r] * FEA + f], msg);
  }
}

// ---------------------------------------------------------------------------
// Kernel 5: per-column sum/sumsq of summed[50000 x 64] for BN2.
// 250 blocks x 200 rows each; consecutive threads -> consecutive cols.
// ---------------------------------------------------------------------------
__global__ __launch_bounds__(256) void bn2_stats(
    const float* __restrict__ summed, float* __restrict__ gstat2) {
  __shared__ float ls[256], lq[256];
  const int tid = threadIdx.x;
  const int col = tid & 63, grp = tid >> 6;   // 4 row-groups
  const int r0  = blockIdx.x * 200;
  float s = 0.0f, q = 0.0f;
  for (int r = r0 + grp; r < r0 + 200; r += 4) {
    float v = summed[(size_t)r * FEA + col];
    s += v; q += v * v;
  }
  ls[tid] = s; lq[tid] = q;
  __syncthreads();
  if (tid < 64) {
    s = ls[tid] + ls[tid + 64] + ls[tid + 128] + ls[tid + 192];
    q = lq[tid] + lq[tid + 64] + lq[tid + 128] + lq[tid + 192];
    atomicAdd(&gstat2[tid], s);
    atomicAdd(&gstat2[FEA + tid], q);
  }
}

__global__ void bn2_finalize(const float* __restrict__ gstat2,
                             const float* __restrict__ g,
                             const float* __restrict__ b,
                             float* __restrict__ out /* scale[64],shift[64] */) {
  const int t = threadIdx.x;  // 64 threads
  const float invN = 1.0f / (float)NNODES;
  float mean = gstat2[t] * invN;
  float var  = gstat2[FEA + t] * invN - mean * mean;
  float sc   = g[t] * rsqrtf(var + BN_EPS);
  out[t]       = sc;
  out[FEA + t] = b[t] - mean * sc;
}

// ---------------------------------------------------------------------------
// Kernel 6: x = softplus(x + BN2(summed))  (elementwise, in place)
// ---------------------------------------------------------------------------
__global__ __launch_bounds__(256) void final_update(
    float* __restrict__ x, const float* __restrict__ summed,
    const float* __restrict__ bn2 /* scale[64],shift[64] */) {
  const int idx = blockIdx.x * 256 + threadIdx.x;
  if (idx < NNODES * FEA) {
    int col = idx & 63;
    float t = summed[idx] * bn2[col] + bn2[FEA + col];
    x[idx] = softplusf_(x[idx] + t);
  }
}

// ---------------------------------------------------------------------------
// Host launch
// ---------------------------------------------------------------------------
extern "C" void kernel_launch(void* const* d_in, const int* in_sizes, int n_in,
                              void* d_out, int out_size, void* d_ws, size_t ws_size,
                              hipStream_t stream) {
  (void)in_sizes; (void)n_in; (void)out_size; (void)ws_size;
  const float* atom  = (const float*)d_in[0];
  const float* nbrf  = (const float*)d_in[1];
  const int*   selfi = (const int*)d_in[2];
  const int*   nbri  = (const int*)d_in[3];
  const float* embW  = (const float*)d_in[4];
  const float* embB  = (const float*)d_in[5];
  const float* fcW   = (const float*)d_in[6];
  const float* fcB   = (const float*)d_in[7];
  const float* bn1g  = (const float*)d_in[8];
  const float* bn1b  = (const float*)d_in[9];
  const float* bn2g  = (const float*)d_in[10];
  const float* bn2b  = (const float*)d_in[11];

  float* x = (float*)d_out;          // working node features == final output
  float* ws      = (float*)d_ws;     // ~12.9 MB total
  float* summed  = ws;                                   // 50000*64
  float* gstat1  = summed + (size_t)NNODES * FEA;        // 256
  float* bn1d    = gstat1 + 2 * N2F;                     // 256
  float* gstat2  = bn1d   + 2 * N2F;                     // 128
  float* bn2d    = gstat2 + 2 * FEA;                     // 128

  embed_kernel<<<NNODES / 16, 128, 0, stream>>>(atom, embW, embB, x);

  for (int l = 0; l < NGRAPH; ++l) {
    hipMemsetAsync(summed, 0, (size_t)NNODES * FEA * sizeof(float), stream);
    hipMemsetAsync(gstat1, 0, 2 * N2F * sizeof(float), stream);
    hipMemsetAsync(gstat2, 0, 2 * FEA * sizeof(float), stream);

    const float* Wl = fcW + (size_t)l * KCAT * N2F;
    const float* bl = fcB + (size_t)l * N2F;

    edge_pass1<<<NEDGES / TILE_M, 256, 0, stream>>>(x, nbrf, selfi, nbri, Wl, bl, gstat1);
    bn1_finalize<<<1, N2F, 0, stream>>>(gstat1, bn1g + l * N2F, bn1b + l * N2F, bn1d);
    edge_pass2<<<NEDGES / TILE_M, 256, 0, stream>>>(x, nbrf, selfi, nbri, Wl, bl, bn1d, summed);
    bn2_stats<<<250, 256, 0, stream>>>(summed, gstat2);
    bn2_finalize<<<1, FEA, 0, stream>>>(gstat2, bn2g + l * FEA, bn2b + l * FEA, bn2d);
    final_update<<<(NNODES * FEA + 255) / 256, 256, 0, stream>>>(x, summed, bn2d);
  }
}